// DynamicMamba_83897891160355
// MI455X (gfx1250) — compile-verified
//
#include <hip/hip_runtime.h>
#include <math.h>

typedef __attribute__((ext_vector_type(2))) float v2f;
typedef __attribute__((ext_vector_type(8))) float v8f;

#define L_SEQ   1024
#define DMODEL  384
#define DINNER  768
#define DX      1536      // 2*DINNER
#define BN      2
#define NSTATE  16
#define DTRANK  24
#define NXROWS  64        // x_dbl rows padded 56 -> 64

// ---------------------------------------------------------------------------
// Kernel 1: in_proj GEMM  xz[b][e][l] = sum_d W[e][d] * H[b][l][d]
// f32 WMMA 16x16x4, one 16x16 tile per wave, K=384 in steps of 4.
// ---------------------------------------------------------------------------
__global__ __launch_bounds__(256) void k_inproj(const float* __restrict__ W,
                                                const float* __restrict__ H,
                                                float* __restrict__ xz) {
  int wave  = blockIdx.x * 8 + (threadIdx.x >> 5);   // 12288 waves
  int lane  = threadIdx.x & 31;
  int laneM = lane & 15, half = lane >> 4;
  int b  = wave / (96 * 64);
  int r  = wave % (96 * 64);
  int e0 = (r / 64) * 16, l0 = (r % 64) * 16;
  const float* Hb   = H + (size_t)b * L_SEQ * DMODEL;
  const float* arow = W  + (size_t)(e0 + laneM) * DMODEL + 2 * half;
  const float* brow = Hb + (size_t)(l0 + laneM) * DMODEL + 2 * half;
  v8f acc = {};
  for (int k = 0; k < DMODEL; k += 4) {
    v2f a, bv;
    a.x  = arow[k];  a.y  = arow[k + 1];
    bv.x = brow[k];  bv.y = brow[k + 1];
    acc = __builtin_amdgcn_wmma_f32_16x16x4_f32(false, a, false, bv,
                                                (short)0, acc, false, false);
  }
  float* out = xz + (size_t)b * DX * L_SEQ;
#pragma unroll
  for (int rr = 0; rr < 8; ++rr)
    out[(size_t)(e0 + rr + 8 * half) * L_SEQ + l0 + laneM] = acc[rr];
}

// ---------------------------------------------------------------------------
// Kernel 2: column sums of xz over l (for cls pooling). One block per (b,e).
// ---------------------------------------------------------------------------
__global__ __launch_bounds__(256) void k_cls(const float* __restrict__ xz,
                                             float* __restrict__ cls) {
  __shared__ float sm[256];
  const float* p = xz + (size_t)blockIdx.x * L_SEQ;
  float s = 0.f;
  for (int t = threadIdx.x; t < L_SEQ; t += 256) s += p[t];
  sm[threadIdx.x] = s;
  __syncthreads();
  for (int st = 128; st > 0; st >>= 1) {
    if ((int)threadIdx.x < st) sm[threadIdx.x] += sm[threadIdx.x + st];
    __syncthreads();
  }
  if (threadIdx.x == 0) cls[blockIdx.x] = sm[0];
}

// ---------------------------------------------------------------------------
// Kernel 3: routing head. Bit-exact JAX threefry2x32 gumbel with key(42).
// policy forward value == y_hard == (logits + g1 - g2 > 0).
// ---------------------------------------------------------------------------
__device__ inline void tf2x32(unsigned k0, unsigned k1, unsigned x0, unsigned x1,
                              unsigned* o0, unsigned* o1) {
  const unsigned r0[4] = {13u, 15u, 26u, 6u}, r1[4] = {17u, 29u, 16u, 24u};
  unsigned ks[3] = {k0, k1, k0 ^ k1 ^ 0x1BD11BDAu};
  x0 += ks[0]; x1 += ks[1];
  for (int i = 1; i <= 5; ++i) {
    const unsigned* R = (i & 1) ? r0 : r1;
#pragma unroll
    for (int j = 0; j < 4; ++j) {
      x0 += x1; x1 = (x1 << R[j]) | (x1 >> (32 - R[j])); x1 ^= x0;
    }
    x0 += ks[i % 3]; x1 += ks[(i + 1) % 3] + (unsigned)i;
  }
  *o0 = x0; *o1 = x1;
}

__device__ inline float u01(unsigned bits) {
  float f = __uint_as_float((bits >> 9) | 0x3f800000u) - 1.0f;
  const float tiny = 1.1754944e-38f;
  float u = f + tiny;
  return u < tiny ? tiny : u;
}

__global__ void k_policy(const float* __restrict__ cls, const float* __restrict__ mask,
                         const float* __restrict__ bhw, const float* __restrict__ bhb,
                         float* __restrict__ msk) {
  if (threadIdx.x != 0 || blockIdx.x != 0) return;
  unsigned a0, b0, a1, b1;
  tf2x32(0u, 42u, 0u, 2u, &a0, &b0);   // split(key(42),2): lanes (0,2),(1,3)
  tf2x32(0u, 42u, 1u, 3u, &a1, &b1);   // kg0=(a0,a1), kg1=(b0,b1)
  unsigned c0, e0, c1, e1, d0, f0, d1, f1;
  tf2x32(a0, a1, 0u, 2u, &c0, &e0);    // g1 bits, shape (2,1,2)
  tf2x32(a0, a1, 1u, 3u, &c1, &e1);
  tf2x32(b0, b1, 0u, 2u, &d0, &f0);    // g2 bits
  tf2x32(b0, b1, 1u, 3u, &d1, &f1);
  unsigned g1b[2][2] = {{c0, c1}, {e0, e1}};
  unsigned g2b[2][2] = {{d0, d1}, {f0, f1}};
  for (int b = 0; b < BN; ++b) {
    float nk = 0.f;
    for (int l = 0; l < L_SEQ; ++l) nk += mask[b * L_SEQ + l];
    float inv = 1.0f / (nk + 1.0f);
    for (int j = 0; j < 2; ++j) {
      float lg = bhb[j];
      const float* wr = bhw + (size_t)j * DX;
      const float* cr = cls + (size_t)b * DX;
      for (int e = 0; e < DX; ++e) lg += (cr[e] * inv) * wr[e];
      float g1 = -logf(-logf(u01(g1b[b][j])));
      float g2 = -logf(-logf(u01(g2b[b][j])));
      msk[b * 2 + j] = ((lg + g1 - g2) > 0.0f) ? 1.0f : 0.0f;
    }
  }
}

// ---------------------------------------------------------------------------
// Kernel 4: depthwise causal conv (k=4) + SiLU, both directions.
// dir==1 reads xz/mask time-reversed; output stored in scan-time order.
// ---------------------------------------------------------------------------
__global__ __launch_bounds__(256) void k_conv(const float* __restrict__ xz,
                                              const float* __restrict__ mask,
                                              const float* __restrict__ wf,
                                              const float* __restrict__ bf,
                                              const float* __restrict__ wb,
                                              const float* __restrict__ bb,
                                              float* __restrict__ conv) {
  int blk = blockIdx.x;                         // dir*BN*DINNER + b*DINNER + d
  int dir = blk / (BN * DINNER);
  int r   = blk % (BN * DINNER);
  int b = r / DINNER, d = r % DINNER;
  const float* w = (dir ? wb : wf) + d * 4;
  float bias = (dir ? bb : bf)[d];
  float w0 = w[0], w1 = w[1], w2 = w[2], w3 = w[3];
  const float* xrow = xz + ((size_t)b * DX + d) * L_SEQ;
  const float* mrow = mask + (size_t)b * L_SEQ;
  float* orow = conv + (((size_t)dir * BN + b) * DINNER + d) * L_SEQ;
  for (int t = threadIdx.x; t < L_SEQ; t += 256) {
    float acc = bias;
#pragma unroll
    for (int j = 0; j < 4; ++j) {
      int p = t - 3 + j;
      float xv = 0.f;
      if (p >= 0) {
        int idx = dir ? (L_SEQ - 1 - p) : p;
        xv = xrow[idx] * mrow[idx];
      }
      acc += (j == 0 ? w0 : j == 1 ? w1 : j == 2 ? w2 : w3) * xv;
    }
    float s = acc / (1.0f + __expf(-acc));      // SiLU
    int idt = dir ? (L_SEQ - 1 - t) : t;
    orow[t] = s * mrow[idt];
  }
}

// ---------------------------------------------------------------------------
// Kernel 5: x_proj GEMM  x_dbl[db][e][t] = sum_d XP[e][d]*conv[db][d][t]
// 56 rows padded to 64 (pad rows use zero A -> zero output).
// ---------------------------------------------------------------------------
__global__ __launch_bounds__(256) void k_xproj(const float* __restrict__ xpf,
                                               const float* __restrict__ xpb,
                                               const float* __restrict__ conv,
                                               float* __restrict__ xdbl) {
  int wave  = blockIdx.x * 8 + (threadIdx.x >> 5);    // 1024 waves
  int lane  = threadIdx.x & 31;
  int laneM = lane & 15, half = lane >> 4;
  int db = wave / 256;                                // dir*BN + b
  int r  = wave % 256;
  int e0 = (r / 64) * 16, l0 = (r % 64) * 16;
  const float* XP = (db >= BN) ? xpb : xpf;           // [56][768]
  const float* CO = conv + (size_t)db * DINNER * L_SEQ;
  int  arow_e = e0 + laneM;
  bool valid  = arow_e < (DTRANK + 2 * NSTATE);
  const float* ap = XP + (size_t)arow_e * DINNER + 2 * half;
  v8f acc = {};
  for (int k = 0; k < DINNER; k += 4) {
    v2f a, bv;
    a.x = valid ? ap[k] : 0.f;
    a.y = valid ? ap[k + 1] : 0.f;
    int kb = k + 2 * half;
    bv.x = CO[(size_t)kb * L_SEQ + l0 + laneM];
    bv.y = CO[(size_t)(kb + 1) * L_SEQ + l0 + laneM];
    acc = __builtin_amdgcn_wmma_f32_16x16x4_f32(false, a, false, bv,
                                                (short)0, acc, false, false);
  }
  float* OUT = xdbl + (size_t)db * NXROWS * L_SEQ;
#pragma unroll
  for (int rr = 0; rr < 8; ++rr)
    OUT[(size_t)(e0 + rr + 8 * half) * L_SEQ + l0 + laneM] = acc[rr];
}

// ---------------------------------------------------------------------------
// Kernel 6: delta = softplus(dt_proj_w @ x_dbl[:24] + dt_bias)   (K=24, VALU)
// ---------------------------------------------------------------------------
__global__ __launch_bounds__(256) void k_delta(const float* __restrict__ dtwf,
                                               const float* __restrict__ dtbf,
                                               const float* __restrict__ dtwb,
                                               const float* __restrict__ dtbb,
                                               const float* __restrict__ xdbl,
                                               float* __restrict__ delta) {
  int blk = blockIdx.x;
  int dir = blk / (BN * DINNER);
  int r   = blk % (BN * DINNER);
  int b = r / DINNER, d = r % DINNER;
  const float* W = (dir ? dtwb : dtwf) + (size_t)d * DTRANK;
  float bias = (dir ? dtbb : dtbf)[d];
  const float* X = xdbl + (size_t)(dir * BN + b) * NXROWS * L_SEQ;
  float* O = delta + (((size_t)dir * BN + b) * DINNER + d) * L_SEQ;
  for (int t = threadIdx.x; t < L_SEQ; t += 256) {
    float acc = bias;
#pragma unroll
    for (int rr = 0; rr < DTRANK; ++rr) acc += W[rr] * X[(size_t)rr * L_SEQ + t];
    O[t] = (acc > 20.f) ? acc : log1pf(__expf(acc));  // softplus
  }
}

// ---------------------------------------------------------------------------
// Kernel 7: selective scan. thread = (d-in-block, n); 1 state/lane;
// Σ_n state*C via 4x shfl_xor inside 16-lane groups (wave32).
// Time axis processed in chunks of 4 with float4 (global_load_b128) loads so
// the per-step serial chain only waits on one vector load per 4 steps.
// ---------------------------------------------------------------------------
__global__ __launch_bounds__(256) void k_scan(const float* __restrict__ Alogf,
                                              const float* __restrict__ Alogb,
                                              const float* __restrict__ Dfp,
                                              const float* __restrict__ Dbp,
                                              const float* __restrict__ xz,
                                              const float* __restrict__ conv,
                                              const float* __restrict__ xdbl,
                                              const float* __restrict__ delta,
                                              float* __restrict__ ysc) {
  int blk = blockIdx.x;                  // dir*BN*48 + b*48 + dblk
  int dir = blk / (BN * 48);
  int r   = blk % (BN * 48);
  int b = r / 48, dblk = r % 48;
  int n = threadIdx.x & 15;
  int d = dblk * 16 + (threadIdx.x >> 4);
  int db = dir * BN + b;
  float An = -__expf((dir ? Alogb : Alogf)[d * NSTATE + n]);
  float Dd = (dir ? Dbp : Dfp)[d];
  const float4* DL = (const float4*)(delta + ((size_t)db * DINNER + d) * L_SEQ);
  const float4* U  = (const float4*)(conv  + ((size_t)db * DINNER + d) * L_SEQ);
  const float4* BV = (const float4*)(xdbl + (size_t)db * NXROWS * L_SEQ +
                                     (size_t)(DTRANK + n) * L_SEQ);
  const float4* CV = (const float4*)(xdbl + (size_t)db * NXROWS * L_SEQ +
                                     (size_t)(DTRANK + NSTATE + n) * L_SEQ);
  const float* Z = xz + ((size_t)b * DX + DINNER + d) * L_SEQ;
  float4* Y = (float4*)(ysc + ((size_t)db * DINNER + d) * L_SEQ);
  float state = 0.f;
  for (int c = 0; c < L_SEQ / 4; ++c) {
    float4 dl4 = DL[c];
    float4 u4  = U[c];
    float4 b4  = BV[c];
    float4 c4  = CV[c];
    float dls[4] = {dl4.x, dl4.y, dl4.z, dl4.w};
    float us[4]  = {u4.x,  u4.y,  u4.z,  u4.w};
    float bs[4]  = {b4.x,  b4.y,  b4.z,  b4.w};
    float cs[4]  = {c4.x,  c4.y,  c4.z,  c4.w};
    float ys[4];
#pragma unroll
    for (int j = 0; j < 4; ++j) {
      float dl = dls[j], u = us[j];
      state = __expf(dl * An) * state + dl * bs[j] * u;
      float part = state * cs[j];
      part += __shfl_xor(part, 1, 32);
      part += __shfl_xor(part, 2, 32);
      part += __shfl_xor(part, 4, 32);
      part += __shfl_xor(part, 8, 32);
      ys[j] = part + u * Dd;
    }
    if (n == 0) {
      int t0 = c * 4;
      float4 yo;
      float yv[4];
#pragma unroll
      for (int j = 0; j < 4; ++j) {
        int lz = dir ? (L_SEQ - 1 - (t0 + j)) : (t0 + j);
        float z = Z[lz];
        yv[j] = ys[j] * (z / (1.0f + __expf(-z)));
      }
      yo.x = yv[0]; yo.y = yv[1]; yo.z = yv[2]; yo.w = yv[3];
      Y[c] = yo;
    }
  }
}

// ---------------------------------------------------------------------------
// Kernel 8: transpose out_proj_w [384][768] -> Wt [768][384]
// ---------------------------------------------------------------------------
__global__ __launch_bounds__(256) void k_transpose(const float* __restrict__ W,
                                                   float* __restrict__ Wt) {
  int i = blockIdx.x * 256 + threadIdx.x;
  if (i >= DMODEL * DINNER) return;
  int o = i / DINNER, d = i % DINNER;
  Wt[(size_t)d * DMODEL + o] = W[i];
}

// ---------------------------------------------------------------------------
// Kernel 9: combine (fmask*y_f + bmask*rev(y_b)) and out_proj GEMM via WMMA.
// out[b][l][o] = sum_d y[b][d][l] * Wt[d][o]
// ---------------------------------------------------------------------------
__global__ __launch_bounds__(256) void k_outproj(const float* __restrict__ ysc,
                                                 const float* __restrict__ Wt,
                                                 const float* __restrict__ msk,
                                                 float* __restrict__ out) {
  int wave  = blockIdx.x * 8 + (threadIdx.x >> 5);   // 3072 waves
  int lane  = threadIdx.x & 31;
  int laneM = lane & 15, half = lane >> 4;
  int b  = wave / (64 * 24);
  int r  = wave % (64 * 24);
  int l0 = (r / 24) * 16, o0 = (r % 24) * 16;
  float fm = msk[b * 2 + 0], bm = msk[b * 2 + 1];
  const float* YF = ysc + (size_t)(0 * BN + b) * DINNER * L_SEQ;
  const float* YB = ysc + (size_t)(1 * BN + b) * DINNER * L_SEQ;
  int l  = l0 + laneM;
  int lr = L_SEQ - 1 - l;
  v8f acc = {};
  for (int k = 0; k < DINNER; k += 4) {
    int ka = k + 2 * half;
    v2f a, bv;
    a.x = YF[(size_t)ka * L_SEQ + l] * fm + YB[(size_t)ka * L_SEQ + lr] * bm;
    a.y = YF[(size_t)(ka + 1) * L_SEQ + l] * fm + YB[(size_t)(ka + 1) * L_SEQ + lr] * bm;
    bv.x = Wt[(size_t)ka * DMODEL + o0 + laneM];
    bv.y = Wt[(size_t)(ka + 1) * DMODEL + o0 + laneM];
    acc = __builtin_amdgcn_wmma_f32_16x16x4_f32(false, a, false, bv,
                                                (short)0, acc, false, false);
  }
  float* O = out + (size_t)b * L_SEQ * DMODEL;
#pragma unroll
  for (int rr = 0; rr < 8; ++rr)
    O[(size_t)(l0 + rr + 8 * half) * DMODEL + o0 + laneM] = acc[rr];
}

// ---------------------------------------------------------------------------
extern "C" void kernel_launch(void* const* d_in, const int* in_sizes, int n_in,
                              void* d_out, int out_size, void* d_ws, size_t ws_size,
                              hipStream_t stream) {
  (void)in_sizes; (void)n_in; (void)out_size; (void)ws_size;
  const float* hidden = (const float*)d_in[0];
  const float* mask   = (const float*)d_in[1];
  const float* inw    = (const float*)d_in[2];
  const float* cw     = (const float*)d_in[3];
  const float* cb     = (const float*)d_in[4];
  const float* xpw    = (const float*)d_in[5];
  const float* dtw    = (const float*)d_in[6];
  const float* dtb    = (const float*)d_in[7];
  const float* alog   = (const float*)d_in[8];
  const float* dsk    = (const float*)d_in[9];
  const float* cwb    = (const float*)d_in[10];
  const float* cbb    = (const float*)d_in[11];
  const float* xpwb   = (const float*)d_in[12];
  const float* dtwb   = (const float*)d_in[13];
  const float* dtbb   = (const float*)d_in[14];
  const float* alogb  = (const float*)d_in[15];
  const float* dskb   = (const float*)d_in[16];
  const float* bhw    = (const float*)d_in[17];
  const float* bhb    = (const float*)d_in[18];
  const float* outw   = (const float*)d_in[19];

  float* ws = (float*)d_ws;
  size_t o_xz    = 0;
  size_t o_cls   = o_xz    + (size_t)BN * DX * L_SEQ;
  size_t o_msk   = o_cls   + (size_t)BN * DX;
  size_t o_conv  = o_msk   + 16;
  size_t o_xdbl  = o_conv  + (size_t)2 * BN * DINNER * L_SEQ;
  size_t o_delta = o_xdbl  + (size_t)2 * BN * NXROWS * L_SEQ;
  size_t o_y     = o_delta + (size_t)2 * BN * DINNER * L_SEQ;
  size_t o_wt    = o_y     + (size_t)2 * BN * DINNER * L_SEQ;
  float* xzb   = ws + o_xz;
  float* clsb  = ws + o_cls;
  float* mskb  = ws + o_msk;
  float* convb = ws + o_conv;
  float* xdblb = ws + o_xdbl;
  float* deltb = ws + o_delta;
  float* yb    = ws + o_y;
  float* wtb   = ws + o_wt;

  k_inproj <<<1536, 256, 0, stream>>>(inw, hidden, xzb);
  k_cls    <<<BN * DX, 256, 0, stream>>>(xzb, clsb);
  k_policy <<<1, 64, 0, stream>>>(clsb, mask, bhw, bhb, mskb);
  k_conv   <<<2 * BN * DINNER, 256, 0, stream>>>(xzb, mask, cw, cb, cwb, cbb, convb);
  k_xproj  <<<128, 256, 0, stream>>>(xpw, xpwb, convb, xdblb);
  k_delta  <<<2 * BN * DINNER, 256, 0, stream>>>(dtw, dtb, dtwb, dtbb, xdblb, deltb);
  k_scan   <<<2 * BN * 48, 256, 0, stream>>>(alog, alogb, dsk, dskb, xzb, convb,
                                             xdblb, deltb, yb);
  k_transpose<<<(DMODEL * DINNER + 255) / 256, 256, 0, stream>>>(outw, wtb);
  k_outproj<<<384, 256, 0, stream>>>(yb, wtb, mskb, (float*)d_out);
}